// Net_4544075399853
// MI455X (gfx1250) — compile-verified
//
#include <hip/hip_runtime.h>
#include <math.h>
#include <stdint.h>

// ---------------------------------------------------------------------------
// LSTM (HIDDEN=20) over T=2048 steps, B=8192 rows, fp32, using CDNA5
// V_WMMA_F32_16X16X4_F32 for the recurrent GEMM.
//
// Per wave: one 16-row batch tile for the entire T loop (512 waves total).
// Gate matrix Wcat[40][128]: K = [hid(20) | h(20)], padded gate columns
// col = q*32 + n  (q = i,f,g,o ; n = hidden unit, cols n>=20 zero).
// With 8 WMMA N-tiles, gate q of (m,n) sits at the SAME lane/VGPR slot in
// accumulator tile t8 = 2q + (n>=16)  -> elementwise LSTM update needs no
// cross-lane data movement.
//
// fp32 WMMA VGPR layouts (CDNA5 ISA 7.12.2):
//   A 16x4 : lane l, vgpr v : M = l&15,         K = 2*(l>>4) + v
//   B 4x16 : lane l, vgpr v : K = 2*(l>>4) + v, N = l&15
//   C 16x16: lane l, vgpr r : M = r + 8*(l>>4), N = l&15
//
// Accumulators start from inline-0 C; gate biases are folded into the
// activation-function constants (zero extra instructions).
// x is prefetched 64 steps ahead with GLOBAL_LOAD_ASYNC_TO_LDS_B128
// (double-buffered; in-order async completion => s_wait_asynccnt 8 means
// "current chunk resident, next chunk still in flight").
// ---------------------------------------------------------------------------

typedef float v2f __attribute__((ext_vector_type(2)));
typedef float v8f __attribute__((ext_vector_type(8)));

static constexpr int Bsz   = 8192;
static constexpr int Tsz   = 2048;
static constexpr int WAVES = 4;

#if defined(__has_builtin)
#if __has_builtin(__builtin_amdgcn_tanhf)
#define USE_HW_TANH 1
#endif
#endif

static constexpr float L2E = 1.4426950408889634f;   // log2(e)

#ifdef USE_HW_TANH
// cb for sig_b is 0.5*bias ; cb for tanh_b is bias
__device__ __forceinline__ float sig_b(float x, float cb) {
  return fmaf(0.5f, __builtin_amdgcn_tanhf(fmaf(x, 0.5f, cb)), 0.5f);
}
__device__ __forceinline__ float tanh_b(float x, float cb) {
  return __builtin_amdgcn_tanhf(x + cb);
}
__device__ __forceinline__ float tanh_c(float x) {
  return __builtin_amdgcn_tanhf(x);
}
#else
// cb for sig_b is -L2E*bias ; cb for tanh_b is -2*L2E*bias
__device__ __forceinline__ float sig_b(float x, float cb) {
  // sigmoid(x+b) = 1/(1+2^(-L2E*x + cb)); 2^large -> inf -> rcp = 0. Safe.
  float t = exp2f(fmaf(x, -L2E, cb));
  return __builtin_amdgcn_rcpf(1.0f + t);
}
__device__ __forceinline__ float tanh_b(float x, float cb) {
  // tanh(x+b) = 2/(1+2^(-2*L2E*x + cb)) - 1; overflow -> -1. Safe.
  float t = exp2f(fmaf(x, -2.0f * L2E, cb));
  return fmaf(2.0f, __builtin_amdgcn_rcpf(1.0f + t), -1.0f);
}
__device__ __forceinline__ float tanh_c(float x) {
  float t = exp2f(x * (-2.0f * L2E));
  return fmaf(2.0f, __builtin_amdgcn_rcpf(1.0f + t), -1.0f);
}
#endif

// Compiler-only reordering barrier. LDS ops from one wave execute in order
// (CDNA5 ISA 7.1: LDS ops are counted by DScnt and return in order), so same-wave cross-lane
// store->load needs no hardware wait; the compiler still inserts precise
// s_wait_dscnt for its own register data dependencies.
__device__ __forceinline__ void lds_compiler_barrier() {
  asm volatile("" ::: "memory");
}

__global__ __launch_bounds__(32 * WAVES)
void lstm_wmma_kernel(const float* __restrict__ x,    // [B,T]
                      const float* __restrict__ W1,   // [20]
                      const float* __restrict__ b1,   // [20]
                      const float* __restrict__ Wih,  // [80][20] (i,f,g,o)
                      const float* __restrict__ Whh,  // [80][20]
                      const float* __restrict__ bih,  // [80]
                      const float* __restrict__ bhh,  // [80]
                      const float* __restrict__ W2,   // [20]
                      const float* __restrict__ b2,   // [1]
                      float* __restrict__ out)        // [B,T]
{
  // B fragments of Wcat, stored in fragment order: [(t8*10+j)*64 + lane*2 + v]
  __shared__ __align__(16) float Wfrag[8 * 10 * 64];           // 20.0 KB
  __shared__ __align__(16) float hbuf[WAVES][16][22];          // h state
  __shared__ __align__(16) float xbuf[WAVES][2][16][68];       // x double-buffer

  const int tid  = threadIdx.x;
  const int lane = tid & 31;
  const int w    = tid >> 5;
  const int half = lane >> 4;   // 0/1
  const int l16  = lane & 15;

  // ---- one-time: build swizzled weight fragments in LDS ----
  for (int idx = tid; idx < 8 * 10 * 64; idx += blockDim.x) {
    const int v  = idx & 1;
    const int l  = (idx >> 1) & 31;
    const int j  = (idx >> 6) % 10;
    const int t8 = idx / 640;
    const int k  = 4 * j + 2 * (l >> 4) + v;   // K row in [0,40)
    const int c  = 16 * t8 + (l & 15);         // padded gate col [0,128)
    const int q  = c >> 5, n = c & 31;
    float val = 0.0f;
    if (n < 20) {
      const int r = q * 20 + n;                // torch gate row
      val = (k < 20) ? Wih[r * 20 + k] : Whh[r * 20 + (k - 20)];
    }
    Wfrag[idx] = val;
  }
  __syncthreads();   // only inter-wave sync in the whole kernel

  // ---- per-lane activation constants with fused gate bias ----
  // this lane's hidden units: n = nt*16 + l16 for nt=0,1
  float cI[2], cF[2], cG[2], cO[2];
#pragma unroll
  for (int nt = 0; nt < 2; ++nt) {
    const int n = nt * 16 + l16;
    float Bi = 0.f, Bf = 0.f, Bg = 0.f, Bo = 0.f;
    if (n < 20) {
      Bi = bih[0 * 20 + n] + bhh[0 * 20 + n];
      Bf = bih[1 * 20 + n] + bhh[1 * 20 + n];
      Bg = bih[2 * 20 + n] + bhh[2 * 20 + n];
      Bo = bih[3 * 20 + n] + bhh[3 * 20 + n];
    }
#ifdef USE_HW_TANH
    cI[nt] = 0.5f * Bi;  cF[nt] = 0.5f * Bf;
    cG[nt] = Bg;         cO[nt] = 0.5f * Bo;
#else
    cI[nt] = -L2E * Bi;  cF[nt] = -L2E * Bf;
    cG[nt] = -2.0f * L2E * Bg;
    cO[nt] = -L2E * Bo;
#endif
  }
  float w1a[5], w1b[5], b1a[5], b1b[5];
#pragma unroll
  for (int j = 0; j < 5; ++j) {
    const int k0 = 4 * j + 2 * half;           // K position this lane feeds
    w1a[j] = W1[k0];  w1b[j] = W1[k0 + 1];
    b1a[j] = b1[k0];  b1b[j] = b1[k0 + 1];
  }
  const float w2a = W2[l16];
  const float w2b = (l16 < 4) ? W2[16 + l16] : 0.0f;
  const float b2v = b2[0];

  const int row0 = (blockIdx.x * WAVES + w) * 16;

  // async prefetch of one 16-row x 64-step x chunk into xbuf[w][buf]
  auto issue_xchunk = [&](int buf, int t0) {
#pragma unroll
    for (int i = 0; i < 8; ++i) {
      const int lin = i * 32 + lane;
      const int rr = lin >> 4, c4 = lin & 15;
      const float* gp = &x[(size_t)(row0 + rr) * Tsz + t0 + c4 * 4];
      const uint32_t lp =
          (uint32_t)(uintptr_t)&xbuf[w][buf][rr][c4 * 4];   // LDS byte offset
      asm volatile("global_load_async_to_lds_b128 %0, %1, off"
                   :: "v"(lp), "v"(gp) : "memory");
    }
  };

  // zero h buffer (cols 0..19 are read)
  for (int i = lane; i < 16 * 22; i += 32)
    hbuf[w][i / 22][i % 22] = 0.0f;

  float cst[16];                               // c state: [nt*8 + r]
#pragma unroll
  for (int i = 0; i < 16; ++i) cst[i] = 0.0f;
  lds_compiler_barrier();

  issue_xchunk(0, 0);                          // chunk 0 in flight

  // ---- sequential time loop ----
  for (int t = 0; t < Tsz; ++t) {
    const int tt = t & 63;
    if (tt == 0) {
      const int c   = t >> 6;
      const int nt0 = (c + 1 < Tsz / 64) ? (t + 64) : t;  // last: dup prefetch
      issue_xchunk((c + 1) & 1, nt0);          // next chunk -> other buffer
      // in-order async completion: <=8 outstanding => current chunk resident
      asm volatile("s_wait_asynccnt 8" ::: "memory");
    }
    const float xv = xbuf[w][(t >> 6) & 1][l16][tt];

    // A fragments: K 0..19 = hid(x), K 20..39 = h (from LDS, A-layout)
    v2f A[10];
#pragma unroll
    for (int j = 0; j < 5; ++j) {
      v2f a;
      a.x = fmaf(xv, w1a[j], b1a[j]);
      a.y = fmaf(xv, w1b[j], b1b[j]);
      A[j] = a;
    }
#pragma unroll
    for (int j = 5; j < 10; ++j) {
      const int f0 = 4 * (j - 5) + 2 * half;   // even -> 8B aligned
      A[j] = *(const v2f*)&hbuf[w][l16][f0];
    }

    // gates = A @ Wcat : 8 N-tiles x 10 K-steps of fp32 WMMA, C starts at
    // inline 0 (bias is folded into the activations below).
    const v8f zero8 = {};
    v8f acc[8];
#pragma unroll
    for (int t8 = 0; t8 < 8; ++t8) {
      const v2f bf0 = *(const v2f*)&Wfrag[(t8 * 10) * 64 + lane * 2];
      v8f a = __builtin_amdgcn_wmma_f32_16x16x4_f32(
                  false, A[0], false, bf0, (short)0, zero8, false, false);
#pragma unroll
      for (int j = 1; j < 10; ++j) {
        const v2f bf = *(const v2f*)&Wfrag[(t8 * 10 + j) * 64 + lane * 2];
        a = __builtin_amdgcn_wmma_f32_16x16x4_f32(
                false, A[j], false, bf, (short)0, a, false, false);
      }
      acc[t8] = a;
    }

    // elementwise LSTM update; i,f,g,o aligned per-lane across acc tiles
    float p[8];
#pragma unroll
    for (int r = 0; r < 8; ++r) p[r] = 0.0f;

#pragma unroll
    for (int nt = 0; nt < 2; ++nt) {
#pragma unroll
      for (int r = 0; r < 8; ++r) {
        const float ig = sig_b (acc[0 + nt][r], cI[nt]);
        const float fg = sig_b (acc[2 + nt][r], cF[nt]);
        const float gg = tanh_b(acc[4 + nt][r], cG[nt]);
        const float og = sig_b (acc[6 + nt][r], cO[nt]);
        const float cn = fg * cst[nt * 8 + r] + ig * gg;
        const float hn = og * tanh_c(cn);
        cst[nt * 8 + r] = cn;
        p[r] = fmaf(hn, nt ? w2b : w2a, p[r]);
        if (nt == 0 || l16 < 4)               // store only valid hidden cols
          hbuf[w][r + 8 * half][nt * 16 + l16] = hn;
      }
    }
    lds_compiler_barrier();   // hw keeps same-wave LDS ops in order

    // out[m,t] = sum_n h[m,n]*W2[n] + b2 : butterfly over the 16-lane half
#pragma unroll
    for (int r = 0; r < 8; ++r) {
      float v = p[r];
      v += __shfl_xor(v, 1, 32);
      v += __shfl_xor(v, 2, 32);
      v += __shfl_xor(v, 4, 32);
      v += __shfl_xor(v, 8, 32);
      p[r] = v;
    }
    if (l16 == 0) {
#pragma unroll
      for (int r = 0; r < 8; ++r)
        out[(size_t)(row0 + r + 8 * half) * Tsz + t] = p[r] + b2v;
    }
  }
}

extern "C" void kernel_launch(void* const* d_in, const int* in_sizes, int n_in,
                              void* d_out, int out_size, void* d_ws, size_t ws_size,
                              hipStream_t stream) {
  const float* x   = (const float*)d_in[0];
  const float* W1  = (const float*)d_in[1];
  const float* b1  = (const float*)d_in[2];
  const float* Wih = (const float*)d_in[3];
  const float* Whh = (const float*)d_in[4];
  const float* bih = (const float*)d_in[5];
  const float* bhh = (const float*)d_in[6];
  const float* W2  = (const float*)d_in[7];
  const float* b2  = (const float*)d_in[8];
  float* out = (float*)d_out;

  const int blocks = Bsz / (16 * WAVES);       // 128 blocks x 128 threads
  lstm_wmma_kernel<<<blocks, 32 * WAVES, 0, stream>>>(
      x, W1, b1, Wih, Whh, bih, bhh, W2, b2, out);
}